// Primitive_70411693850872
// MI455X (gfx1250) — compile-verified
//
#include <hip/hip_runtime.h>
#include <hip/hip_bf16.h>

// CDNA5 / gfx1250, wave32.
// Apply 2x2 gate to qubit axis TARGET=5 of a (2,)^24 x (4,) f32 state.
//
// element index = hi * 2^21 + t * 2^20 + lo ; pair stride TSTRIDE = 2^20 elems.
// 2^25 pairs. Bandwidth bound: 512 MB @ 23.3 TB/s => ~22 us floor; WMMA is the
// free ALU path (V_WMMA_F32_16X16X4_F32, D = A*B + C) with a layout-natural
// packing:
//   A (16x4): two diagonal 2x2 gate blocks (rows 0-1/cols 0-1, rows 2-3/cols 2-3)
//   B (4x16): col n = (s0[n], s1[n], s0[n+16], s1[n+16])
//   ISA B striping (VGPR0 = K0|K2 across lane halves, VGPR1 = K1|K3) means
//   b.x / b.y are PLAIN coalesced 32-lane loads of s0 / s1 -- zero input shuffles.
//   D rows 0/1 (sites 0-15) and 2/3 (sites 16-31) live in lanes 0-15 of d[0..3];
//   two bpermute+select repack them into full 128B coalesced stores.
//
// MLP: 4 tiles (128 pairs, 1 KB) per wave iteration -- 8 loads clause-issued
// before the first WMMA waits, so each wave keeps 1 KB in flight.

typedef __attribute__((ext_vector_type(2))) float v2f;
typedef __attribute__((ext_vector_type(8))) float v8f;

#define TSTRIDE  (1u << 20)          // element stride along target axis
#define NPAIRS   (1u << 25)
#define NGROUPS  (NPAIRS / 128u)     // 128 pairs (4 WMMA tiles) per group -> 2^18

__global__ __launch_bounds__(256) void pauli_gate_wmma(
    const float* __restrict__ state,
    const float* __restrict__ pauli,
    float* __restrict__ out)
{
    const unsigned lane        = threadIdx.x & 31u;
    const unsigned waveInBlk   = threadIdx.x >> 5;
    const unsigned wavesPerBlk = blockDim.x >> 5;
    const unsigned gwave       = blockIdx.x * wavesPerBlk + waveInBlk;
    const unsigned nwaves      = gridDim.x * wavesPerBlk;

    const float p00 = pauli[0], p01 = pauli[1], p10 = pauli[2], p11 = pauli[3];

    // A per ISA striping: VGPR0 = K=0 (lanes 0-15, M=lane) | K=2 (lanes 16-31, M=lane-16)
    //                     VGPR1 = K=1 | K=3
    const bool row0 = (lane == 0u) || (lane == 18u);   // M=0 (K0,K1) / M=2 (K2,K3)
    const bool row1 = (lane == 1u) || (lane == 19u);   // M=1 / M=3
    v2f a;
    a.x = row0 ? p00 : (row1 ? p10 : 0.0f);
    a.y = row0 ? p01 : (row1 ? p11 : 0.0f);

    for (unsigned g = gwave; g < NGROUPS; g += nwaves) {
        const unsigned pairBase = g << 7;                    // 128 consecutive pairs
        const unsigned hi       = pairBase >> 20;
        const unsigned lo       = pairBase & (TSTRIDE - 1u); // groups never straddle hi
        const unsigned base     = (hi << 21) + lo + lane;

        // 8 streaming loads issued as one clause before any consumer waits.
        v2f b[4];
#pragma unroll
        for (int u = 0; u < 4; ++u) {
            b[u].x = __builtin_nontemporal_load(state + base + 32u * u);            // s0
            b[u].y = __builtin_nontemporal_load(state + base + 32u * u + TSTRIDE);  // s1
        }

        v8f d[4];
#pragma unroll
        for (int u = 0; u < 4; ++u) {
            v8f c = {};
            d[u] = __builtin_amdgcn_wmma_f32_16x16x4_f32(
                /*neg_a=*/false, a, /*neg_b=*/false, b[u],
                /*c_mod=*/(short)0, c, /*reuse_a=*/false, /*reuse_b=*/false);
        }

#pragma unroll
        for (int u = 0; u < 4; ++u) {
            // d[u][0]/d[u][1] lanes 0-15: out0/out1 sites 0-15;
            // d[u][2]/d[u][3] lanes 0-15: sites 16-31. Repack -> lane L holds site L.
            const float u2 = __shfl(d[u][2], (int)(lane & 15u), 32);
            const float u3 = __shfl(d[u][3], (int)(lane & 15u), 32);
            const float o0 = (lane < 16u) ? d[u][0] : u2;
            const float o1 = (lane < 16u) ? d[u][1] : u3;
            __builtin_nontemporal_store(o0, out + base + 32u * u);
            __builtin_nontemporal_store(o1, out + base + 32u * u + TSTRIDE);
        }
    }
}

extern "C" void kernel_launch(void* const* d_in, const int* in_sizes, int n_in,
                              void* d_out, int out_size, void* d_ws, size_t ws_size,
                              hipStream_t stream) {
    (void)in_sizes; (void)n_in; (void)out_size; (void)d_ws; (void)ws_size;
    const float* state = (const float*)d_in[0];
    const float* pauli = (const float*)d_in[1];
    float*       out   = (float*)d_out;

    // 4096 blocks x 8 waves = 32768 waves; 2^18 groups -> exactly 8 iterations
    // per wave (uniform trip count -> EXEC all-ones around every WMMA).
    dim3 grid(4096), block(256);
    hipLaunchKernelGGL(pauli_gate_wmma, grid, block, 0, stream, state, pauli, out);
}